// VectorQuantizer_55542517072479
// MI455X (gfx1250) — compile-verified
//
#include <hip/hip_runtime.h>

typedef float v2f __attribute__((ext_vector_type(2)));
typedef float v8f __attribute__((ext_vector_type(8)));

#define BVAL 32
#define CCH 32
#define HH 16
#define VCODES 8192
#define ELEMS (BVAL * CCH * HH * HH)   // 262144

// ---------------- helpers ----------------
__device__ __forceinline__ float cubic_w(float d) {
    const float A = -0.75f;
    float d2 = d * d, d3 = d2 * d;
    if (d <= 1.f) return (A + 2.f) * d3 - (A + 3.f) * d2 + 1.f;
    if (d < 2.f)  return A * d3 - 5.f * A * d2 + 8.f * A * d - 4.f * A;
    return 0.f;
}

// ---------------- K0: init state + |e|^2 ----------------
__global__ void k_init(const float* __restrict__ f, const float* __restrict__ emb,
                       float* __restrict__ f_rest, float* __restrict__ f_hat,
                       float* __restrict__ emb_sq, float* __restrict__ loss_acc) {
    int i = blockIdx.x * 256 + threadIdx.x;
    if (i < ELEMS) { f_rest[i] = f[i]; f_hat[i] = 0.f; }
    if (i < VCODES) {
        const float* row = emb + i * CCH;
        float s = 0.f;
        #pragma unroll
        for (int k = 0; k < CCH; k++) s = fmaf(row[k], row[k], s);
        emb_sq[i] = s;
    }
    if (i == 0) *loss_acc = 0.f;
}

// ---------------- K1: area pool f_rest -> z [N, C] ----------------
__global__ void k_pool(const float* __restrict__ f_rest, float* __restrict__ z, int pn) {
    int t = blockIdx.x * 256 + threadIdx.x;
    int total = BVAL * pn * pn * CCH;
    if (t >= total) return;
    int c = t & (CCH - 1);
    int n = t >> 5;
    int pnpn = pn * pn;
    int b = n / pnpn;
    int rem = n - b * pnpn;
    int p = rem / pn, q = rem - p * pn;
    int y0 = (p * HH) / pn, y1 = ((p + 1) * HH + pn - 1) / pn;
    int x0 = (q * HH) / pn, x1 = ((q + 1) * HH + pn - 1) / pn;
    const float* base = f_rest + (b * CCH + c) * HH * HH;
    float s = 0.f;
    for (int y = y0; y < y1; y++)
        for (int x = x0; x < x1; x++)
            s += base[y * HH + x];
    z[n * CCH + c] = s / (float)((y1 - y0) * (x1 - x0));
}

// ---------------- K2: WMMA fp32 nearest-code search, 2-way M-blocked ----------------
// One wave owns TWO 16-row tiles of z (N is always a multiple of 32). Each emb tile
// (8 x b64 loads) feeds 16 WMMAs across two independent accumulator chains, doubling
// arithmetic intensity and removing the single dependent-WMMA chain.
// score = |e|^2 - 2 z.e  (|z|^2 dropped: constant per row, argmin-invariant).
// D layout: VGPR r -> rows {r, r+8}; lane&15 -> column; lane>>4 selects row half.
__global__ __launch_bounds__(256) void k_argmin(const float* __restrict__ z,
                                                const float* __restrict__ emb,
                                                const float* __restrict__ emb_sq,
                                                int* __restrict__ idx_out, int numUnits) {
    int wave = threadIdx.x >> 5;
    int lane = threadIdx.x & 31;
    int unit = blockIdx.x * 8 + wave;          // one unit = 32 rows = two M tiles
    if (unit >= numUnits) return;              // wave-uniform: EXEC all-ones for WMMA
    int ml = lane & 15;
    int hi = lane >> 4;

    // A fragments (16x4 f32): lanes 0-15 hold K={k0,k0+1}, lanes 16-31 hold K={k0+2,k0+3}
    int row0 = unit * 32 + ml;
    int row1 = row0 + 16;
    const float2* z2 = reinterpret_cast<const float2*>(z);
    v2f a0[8], a1[8];
    #pragma unroll
    for (int k = 0; k < 8; k++) {
        float2 t0 = z2[row0 * 16 + 2 * k + hi];
        float2 t1 = z2[row1 * 16 + 2 * k + hi];
        a0[k].x = t0.x; a0[k].y = t0.y;
        a1[k].x = t1.x; a1[k].y = t1.y;
    }

    float best0[8], best1[8];
    int   bidx0[8], bidx1[8];
    #pragma unroll
    for (int r = 0; r < 8; r++) {
        best0[r] = 3.4e38f; bidx0[r] = 0;
        best1[r] = 3.4e38f; bidx1[r] = 0;
    }

    const float2* e2 = reinterpret_cast<const float2*>(emb);
    for (int t0c = 0; t0c < VCODES; t0c += 16) {
        int erow = t0c + ml;                   // candidate code this lane's column scores
        v8f acc0 = {0.f, 0.f, 0.f, 0.f, 0.f, 0.f, 0.f, 0.f};
        v8f acc1 = {0.f, 0.f, 0.f, 0.f, 0.f, 0.f, 0.f, 0.f};
        #pragma unroll
        for (int k = 0; k < 8; k++) {          // K = 32 in 8 steps of 4
            float2 e = e2[erow * 16 + 2 * k + hi];
            v2f bf; bf.x = e.x; bf.y = e.y;
            acc0 = __builtin_amdgcn_wmma_f32_16x16x4_f32(
                false, a0[k], false, bf, (short)0, acc0, false, false);
            acc1 = __builtin_amdgcn_wmma_f32_16x16x4_f32(
                false, a1[k], false, bf, (short)0, acc1, false, false);
        }
        float esq = emb_sq[erow];
        #pragma unroll
        for (int r = 0; r < 8; r++) {
            float s0 = fmaf(-2.f, acc0[r], esq);
            float s1 = fmaf(-2.f, acc1[r], esq);
            if (s0 < best0[r]) { best0[r] = s0; bidx0[r] = erow; }  // lowest idx on tie
            if (s1 < best1[r]) { best1[r] = s1; bidx1[r] = erow; }
        }
    }

    // argmin across the 16 lanes of each half (xor masks 1..8 stay within halves)
    #pragma unroll
    for (int r = 0; r < 8; r++) {
        float v0 = best0[r]; int id0 = bidx0[r];
        float v1 = best1[r]; int id1 = bidx1[r];
        #pragma unroll
        for (int m = 1; m < 16; m <<= 1) {
            float ov0 = __shfl_xor(v0, m, 32);
            int   oi0 = __shfl_xor(id0, m, 32);
            float ov1 = __shfl_xor(v1, m, 32);
            int   oi1 = __shfl_xor(id1, m, 32);
            if (ov0 < v0 || (ov0 == v0 && oi0 < id0)) { v0 = ov0; id0 = oi0; }
            if (ov1 < v1 || (ov1 == v1 && oi1 < id1)) { v1 = ov1; id1 = oi1; }
        }
        bidx0[r] = id0; bidx1[r] = id1;
    }
    if (ml == 0) {
        #pragma unroll
        for (int r = 0; r < 8; r++) {
            idx_out[unit * 32 + hi * 8 + r]      = bidx0[r];
            idx_out[unit * 32 + 16 + hi * 8 + r] = bidx1[r];
        }
    }
}

// ---------------- K3: gather codes + bicubic upsample ----------------
// pn==16 degenerates to exact identity weights {0,1,0,0} -> pure gather (matches reference).
__global__ void k_upsample(const float* __restrict__ emb, const int* __restrict__ idx,
                           float* __restrict__ h_up, int pn) {
    int e = blockIdx.x * 256 + threadIdx.x;
    if (e >= ELEMS) return;
    int j = e & 15, i = (e >> 4) & 15, c = (e >> 8) & 31, b = e >> 13;
    float scale = (float)pn / 16.f;
    float sy = ((float)i + 0.5f) * scale - 0.5f;
    float sx = ((float)j + 0.5f) * scale - 0.5f;
    int fy = (int)floorf(sy), fx = (int)floorf(sx);
    float wy[4], wx[4];
    int py[4], px[4];
    #pragma unroll
    for (int k = 0; k < 4; k++) {
        int ty = fy - 1 + k, tx = fx - 1 + k;
        wy[k] = cubic_w(fabsf(sy - (float)ty));
        wx[k] = cubic_w(fabsf(sx - (float)tx));
        py[k] = min(max(ty, 0), pn - 1);
        px[k] = min(max(tx, 0), pn - 1);
    }
    int nb = b * pn * pn;
    float s = 0.f;
    #pragma unroll
    for (int ky = 0; ky < 4; ky++) {
        float rowacc = 0.f;
        #pragma unroll
        for (int kx = 0; kx < 4; kx++) {
            int code = idx[nb + py[ky] * pn + px[kx]];
            rowacc = fmaf(wx[kx], emb[code * CCH + c], rowacc);
        }
        s = fmaf(wy[ky], rowacc, s);
    }
    h_up[e] = s;
}

// ---------------- K4: Phi (3x3 conv residual) + state update + loss partials ----------------
// One block per (b, c_out) plane; all 32 input planes + weights staged in LDS.
__global__ __launch_bounds__(256) void k_phi_update(
        const float* __restrict__ h_up, const float* __restrict__ conv_w,
        const float* __restrict__ conv_b, const float* __restrict__ f_in,
        float* __restrict__ f_hat, float* __restrict__ f_rest,
        float* __restrict__ blockSums, int pi) {
    __shared__ float sH[CCH][HH * HH];   // 32 KB
    __shared__ float sW[CCH * 9];
    __shared__ float sRed[256];
    int bid = blockIdx.x;
    int b = bid >> 5, c = bid & 31;
    int tid = threadIdx.x;

    const float* hin = h_up + b * CCH * HH * HH;
    for (int t = tid; t < CCH * HH * HH; t += 256)
        sH[t >> 8][t & 255] = hin[t];
    const float* wbase = conv_w + (pi * CCH + c) * CCH * 9;
    for (int t = tid; t < CCH * 9; t += 256) sW[t] = wbase[t];
    __syncthreads();

    int i = tid >> 4, j = tid & 15;
    float conv = conv_b[pi * CCH + c];
    for (int ci = 0; ci < CCH; ci++) {
        const float* wr = &sW[ci * 9];
        const float* hp = &sH[ci][0];
        #pragma unroll
        for (int dy = 0; dy < 3; dy++) {
            int y = i + dy - 1;
            if ((unsigned)y < 16u) {
                #pragma unroll
                for (int dx = 0; dx < 3; dx++) {
                    int x = j + dx - 1;
                    if ((unsigned)x < 16u)
                        conv = fmaf(wr[dy * 3 + dx], hp[y * 16 + x], conv);
                }
            }
        }
    }
    int e = bid * 256 + tid;                      // == ((b*32+c)*16+i)*16+j
    float h2 = 0.5f * sH[c][tid] + 0.5f * conv;   // RESI = 0.5
    float nf = f_hat[e] + h2;
    f_hat[e] = nf;
    f_rest[e] -= h2;
    float d = nf - f_in[e];
    sRed[tid] = d * d;
    __syncthreads();
    for (int s = 128; s > 0; s >>= 1) {
        if (tid < s) sRed[tid] += sRed[tid + s];
        __syncthreads();
    }
    if (tid == 0) blockSums[bid] = sRed[0];
}

// ---------------- K5: deterministic loss reduction (fixed tree order) ----------------
__global__ void k_loss_reduce(const float* __restrict__ blockSums, float* __restrict__ loss_acc) {
    __shared__ float sRed[256];
    int tid = threadIdx.x;
    sRed[tid] = blockSums[tid] + blockSums[tid + 256] + blockSums[tid + 512] + blockSums[tid + 768];
    __syncthreads();
    for (int s = 128; s > 0; s >>= 1) {
        if (tid < s) sRed[tid] += sRed[tid + s];
        __syncthreads();
    }
    if (tid == 0) *loss_acc += sRed[0];
}

// ---------------- K6: emit outputs ----------------
__global__ void k_finalize(const float* __restrict__ f_hat, const float* __restrict__ loss_acc,
                           float* __restrict__ out) {
    int i = blockIdx.x * 256 + threadIdx.x;
    if (i < ELEMS) {
        out[i] = f_hat[i];        // straight-through: sg(f_hat) - f + f == f_hat numerically
    } else if (i == ELEMS) {
        const float coef = 1.25f / (10.f * (float)ELEMS);   // (1+BETA)/(SN*B*C*H*W)
        out[ELEMS] = (*loss_acc) * coef;
    }
}

// ---------------- host orchestration ----------------
extern "C" void kernel_launch(void* const* d_in, const int* in_sizes, int n_in,
                              void* d_out, int out_size, void* d_ws, size_t ws_size,
                              hipStream_t stream) {
    (void)in_sizes; (void)n_in; (void)out_size; (void)ws_size;
    const float* f     = (const float*)d_in[0];
    const float* emb   = (const float*)d_in[1];
    const float* convw = (const float*)d_in[2];
    const float* convb = (const float*)d_in[3];
    float* out = (float*)d_out;

    char* ws = (char*)d_ws;
    float* f_rest    = (float*)ws;  ws += (size_t)ELEMS * sizeof(float);
    float* f_hat     = (float*)ws;  ws += (size_t)ELEMS * sizeof(float);
    float* zbuf      = (float*)ws;  ws += (size_t)ELEMS * sizeof(float);
    float* h_up      = (float*)ws;  ws += (size_t)ELEMS * sizeof(float);
    float* emb_sq    = (float*)ws;  ws += (size_t)VCODES * sizeof(float);
    int*   idxbuf    = (int*)ws;    ws += (size_t)VCODES * sizeof(int);
    float* blockSums = (float*)ws;  ws += 1024 * sizeof(float);
    float* loss_acc  = (float*)ws;  ws += sizeof(float);

    static const int PNS[10] = {1, 2, 3, 4, 5, 6, 8, 10, 13, 16};
    static const int PHI[10] = {0, 0, 0, 1, 1, 2, 2, 2, 3, 3};  // shared-Phi tick mapping

    k_init<<<ELEMS / 256, 256, 0, stream>>>(f, emb, f_rest, f_hat, emb_sq, loss_acc);

    for (int si = 0; si < 10; si++) {
        int pn = PNS[si];
        int N = BVAL * pn * pn;            // always a multiple of 32
        int total = N * CCH;
        k_pool<<<(total + 255) / 256, 256, 0, stream>>>(f_rest, zbuf, pn);
        int units = N / 32;                // two 16-row M tiles per wave
        k_argmin<<<(units + 7) / 8, 256, 0, stream>>>(zbuf, emb, emb_sq, idxbuf, units);
        k_upsample<<<ELEMS / 256, 256, 0, stream>>>(emb, idxbuf, h_up, pn);
        k_phi_update<<<BVAL * CCH, 256, 0, stream>>>(h_up, convw, convb, f,
                                                     f_hat, f_rest, blockSums, PHI[si]);
        k_loss_reduce<<<1, 256, 0, stream>>>(blockSums, loss_acc);
    }
    k_finalize<<<(ELEMS + 1 + 255) / 256, 256, 0, stream>>>(f_hat, loss_acc, out);
}